// GAT_79559974191356
// MI455X (gfx1250) — compile-verified
//
#include <hip/hip_runtime.h>
#include <hip/hip_bf16.h>
#include <math.h>

// ---------------------------------------------------------------------------
// GAT (3 layers, 4 heads x 64) + projection, fp32 end-to-end.
// GEMMs use V_WMMA_F32_16X16X4_F32 with A-fragment reuse across CT column
// tiles per wave; B-fragment loads are hoisted ahead of the WMMA group so
// the compiler can clause the loads and issue the 4 WMMAs back-to-back.
// The edge softmax/aggregate phase is memory-bound and uses L2-resident
// buffers + f32 atomics.
// ---------------------------------------------------------------------------

typedef __attribute__((ext_vector_type(2))) float v2f;
typedef __attribute__((ext_vector_type(8))) float v8f;

#define HEADS 4
#define HID   64
#define HC    256
#define NEG_SLOPE 0.2f

// ---------------- WMMA fp32 GEMM: C[M,Ncols] = A[M,K] @ B[K,Ncols] (+bias) --
// grid = (M/16, Ncols/(16*CT)), block = 32 (one wave; 16 x 16*CT strip).
// A frag (16x4 f32): lane<16 -> K={k,k+1}, lane>=16 -> K={k+2,k+3}, M=lane&15.
// B frag (4x16 f32): mirrored K-major. C/D: VGPR r -> row r (+8 for hi half).
template <int CT>
__global__ void __launch_bounds__(32)
gat_wmma_gemm(const float* __restrict__ A, const float* __restrict__ B,
              float* __restrict__ C, int K, int Ncols,
              const float* __restrict__ bias) {
  const int lane = threadIdx.x;
  const int row_base = blockIdx.x * 16;
  const int col_base = blockIdx.y * (16 * CT);
  const int half = lane >> 4;   // 0 or 1
  const int l15  = lane & 15;

  v8f acc[CT];
#pragma unroll
  for (int c = 0; c < CT; ++c) acc[c] = (v8f){};

  const float* Ap = A + (size_t)(row_base + l15) * K + half * 2;
  const float* Bp = B + (size_t)(half * 2) * Ncols + col_base + l15;

  for (int k = 0; k < K; k += 4) {
    const float* Bk = Bp + (size_t)k * Ncols;
    // hoist all loads for this k-step so they can issue as one clause
    v2f a, b[CT];
    a.x = Ap[k];
    a.y = Ap[k + 1];
#pragma unroll
    for (int c = 0; c < CT; ++c) {
      b[c].x = Bk[c * 16];
      b[c].y = Bk[c * 16 + Ncols];
    }
#pragma unroll
    for (int c = 0; c < CT; ++c) {
      acc[c] = __builtin_amdgcn_wmma_f32_16x16x4_f32(
          /*neg_a=*/false, a, /*neg_b=*/false, b[c],
          /*c_mod=*/(short)0, acc[c], /*reuse_a=*/false, /*reuse_b=*/false);
    }
  }

  const int row = row_base + half * 8;
#pragma unroll
  for (int c = 0; c < CT; ++c) {
    const int col = col_base + c * 16 + l15;
    float* Cp = C + (size_t)row * Ncols + col;
    const float bv = bias ? bias[col] : 0.0f;
#pragma unroll
    for (int r = 0; r < 8; ++r) {
      Cp[(size_t)r * Ncols] = acc[c][r] + bv;
    }
  }
}

// ---------------- per-node attention logits --------------------------------
// one wave per node; lane covers 8 consecutive channels; head = lane>>3.
__global__ void gat_alpha(const float* __restrict__ H,
                          const float* __restrict__ a_src,
                          const float* __restrict__ a_dst,
                          float* __restrict__ AS, float* __restrict__ AD,
                          int n) {
  const int wave = blockIdx.x * (blockDim.x >> 5) + (threadIdx.x >> 5);
  if (wave >= n) return;
  const int lane = threadIdx.x & 31;
  const int head = lane >> 3;
  const int coff = (lane & 7) * 8;

  const float* hp  = H + (size_t)wave * HC + lane * 8;
  const float* asp = a_src + head * HID + coff;
  const float* adp = a_dst + head * HID + coff;

  float ss = 0.f, sd = 0.f;
#pragma unroll
  for (int i = 0; i < 8; ++i) {
    float hv = hp[i];
    ss += hv * asp[i];
    sd += hv * adp[i];
  }
#pragma unroll
  for (int off = 1; off < 8; off <<= 1) {
    ss += __shfl_xor(ss, off);
    sd += __shfl_xor(sd, off);
  }
  if ((lane & 7) == 0) {
    AS[wave * HEADS + head] = ss;
    AD[wave * HEADS + head] = sd;
  }
}

// ---------------- init accumulators ---------------------------------------
__global__ void gat_init(float* __restrict__ AGG, float* __restrict__ NMAX,
                         float* __restrict__ DEN, size_t nAgg, int n4) {
  const size_t i = (size_t)blockIdx.x * blockDim.x + threadIdx.x;
  if (i < nAgg) AGG[i] = 0.0f;
  if (i < (size_t)n4) {
    NMAX[i] = -INFINITY;
    DEN[i]  = 0.0f;
  }
}

__device__ __forceinline__ void atomicMaxF(float* addr, float v) {
  if (v >= 0.0f)
    atomicMax((int*)addr, __float_as_int(v));
  else
    atomicMin((unsigned int*)addr, __float_as_uint(v));
}

__device__ __forceinline__ void edge_endpoints(const int* __restrict__ ei,
                                               int E, int e, int& src, int& dst) {
  if (e < E) { src = ei[e]; dst = ei[E + e]; }
  else       { src = dst = e - E; }   // self-loops appended
}

// ---------------- edge scores + segment max --------------------------------
__global__ void gat_edge_score_max(const int* __restrict__ ei, int E, int Etot,
                                   const float* __restrict__ AS,
                                   const float* __restrict__ AD,
                                   float* __restrict__ EE,
                                   float* __restrict__ NMAX) {
  const int t = blockIdx.x * blockDim.x + threadIdx.x;
  if (t >= Etot * HEADS) return;
  const int e = t >> 2, h = t & 3;
  int src, dst;
  edge_endpoints(ei, E, e, src, dst);
  float v = AS[src * HEADS + h] + AD[dst * HEADS + h];
  v = (v > 0.0f) ? v : NEG_SLOPE * v;          // leaky_relu
  EE[t] = v;
  atomicMaxF(&NMAX[dst * HEADS + h], v);
}

// ---------------- exp(e - max) + segment sum -------------------------------
__global__ void gat_edge_exp_sum(const int* __restrict__ ei, int E, int Etot,
                                 float* __restrict__ EE,
                                 const float* __restrict__ NMAX,
                                 float* __restrict__ DEN) {
  const int t = blockIdx.x * blockDim.x + threadIdx.x;
  if (t >= Etot * HEADS) return;
  const int e = t >> 2, h = t & 3;
  int src, dst;
  edge_endpoints(ei, E, e, src, dst);
  const float v = expf(EE[t] - NMAX[dst * HEADS + h]);
  EE[t] = v;
  atomicAdd(&DEN[dst * HEADS + h], v);
}

// ---------------- weighted scatter-aggregate -------------------------------
// one wave per edge; lane handles 8 channels (2x float4 gather, 8 atomics).
__global__ void gat_edge_aggregate(const int* __restrict__ ei, int E, int Etot,
                                   const float* __restrict__ H,
                                   const float* __restrict__ EE,
                                   const float* __restrict__ DEN,
                                   float* __restrict__ AGG) {
  const int wave = blockIdx.x * (blockDim.x >> 5) + (threadIdx.x >> 5);
  if (wave >= Etot) return;
  const int lane = threadIdx.x & 31;
  int src, dst;
  edge_endpoints(ei, E, wave, src, dst);
  const int head = lane >> 3;
  const float alpha = EE[wave * HEADS + head] / DEN[dst * HEADS + head];

  const float4* hp = (const float4*)(H + (size_t)src * HC + lane * 8);
  const float4 h0 = hp[0];
  const float4 h1 = hp[1];
  float* op = AGG + (size_t)dst * HC + lane * 8;
  atomicAdd(op + 0, h0.x * alpha);
  atomicAdd(op + 1, h0.y * alpha);
  atomicAdd(op + 2, h0.z * alpha);
  atomicAdd(op + 3, h0.w * alpha);
  atomicAdd(op + 4, h1.x * alpha);
  atomicAdd(op + 5, h1.y * alpha);
  atomicAdd(op + 6, h1.z * alpha);
  atomicAdd(op + 7, h1.w * alpha);
}

// ---------------- bias + exact GELU ---------------------------------------
__global__ void gat_bias_gelu(const float* __restrict__ AGG,
                              const float* __restrict__ b,
                              float* __restrict__ ACT, size_t total) {
  const size_t i = (size_t)blockIdx.x * blockDim.x + threadIdx.x;
  if (i >= total) return;
  const float v = AGG[i] + b[i & (HC - 1)];
  ACT[i] = 0.5f * v * (1.0f + erff(v * 0.70710678118654752f));
}

// ---------------------------------------------------------------------------
extern "C" void kernel_launch(void* const* d_in, const int* in_sizes, int n_in,
                              void* d_out, int out_size, void* d_ws, size_t ws_size,
                              hipStream_t stream) {
  const float* x  = (const float*)d_in[0];
  const int*   ei = (const int*)d_in[1];
  const float* W[3]  = {(const float*)d_in[2],  (const float*)d_in[6],  (const float*)d_in[10]};
  const float* As[3] = {(const float*)d_in[3],  (const float*)d_in[7],  (const float*)d_in[11]};
  const float* Ad[3] = {(const float*)d_in[4],  (const float*)d_in[8],  (const float*)d_in[12]};
  const float* Bb[3] = {(const float*)d_in[5],  (const float*)d_in[9],  (const float*)d_in[13]};
  const float* Wp = (const float*)d_in[14];
  const float* bp = (const float*)d_in[15];

  const int N    = in_sizes[0] / 128;   // 20000
  const int E    = in_sizes[1] / 2;     // 320000
  const int Etot = E + N;               // +self loops

  // workspace layout (floats)
  float* ws = (float*)d_ws;
  size_t o = 0;
  float* H    = ws + o; o += (size_t)N * HC;     // gemm output
  float* AGG  = ws + o; o += (size_t)N * HC;     // aggregation accumulator
  float* ACT  = ws + o; o += (size_t)N * HC;     // gelu output / next input
  float* EE   = ws + o; o += (size_t)Etot * HEADS;
  float* AS   = ws + o; o += (size_t)N * HEADS;
  float* ADv  = ws + o; o += (size_t)N * HEADS;
  float* NM   = ws + o; o += (size_t)N * HEADS;
  float* DEN  = ws + o; o += (size_t)N * HEADS;

  const size_t nAgg = (size_t)N * HC;
  const int    n4   = N * HEADS;
  const int    eh   = Etot * HEADS;

  const float* cur = x;
  int K = 128;
  for (int l = 0; l < 3; ++l) {
    dim3 gg(N / 16, HC / 64);          // CT=4 -> 64 cols per wave
    gat_wmma_gemm<4><<<gg, 32, 0, stream>>>(cur, W[l], H, K, HC, nullptr);
    gat_alpha<<<(N + 7) / 8, 256, 0, stream>>>(H, As[l], Ad[l], AS, ADv, N);
    gat_init<<<(int)((nAgg + 255) / 256), 256, 0, stream>>>(AGG, NM, DEN, nAgg, n4);
    gat_edge_score_max<<<(eh + 255) / 256, 256, 0, stream>>>(ei, E, Etot, AS, ADv, EE, NM);
    gat_edge_exp_sum<<<(eh + 255) / 256, 256, 0, stream>>>(ei, E, Etot, EE, NM, DEN);
    gat_edge_aggregate<<<(Etot + 7) / 8, 256, 0, stream>>>(ei, E, Etot, H, EE, DEN, AGG);
    gat_bias_gelu<<<(int)((nAgg + 255) / 256), 256, 0, stream>>>(AGG, Bb[l], ACT, nAgg);
    cur = ACT;
    K = HC;
  }

  // final projection [N,256] @ [256,32] + bp -> d_out  (CT=2 -> 32 cols)
  dim3 pg(N / 16, 1);
  gat_wmma_gemm<2><<<pg, 32, 0, stream>>>(ACT, Wp, (float*)d_out, HC, 32, bp);
}